// GATEncoder_46866683134378
// MI455X (gfx1250) — compile-verified
//
#include <hip/hip_runtime.h>
#include <math.h>

typedef float v2f __attribute__((ext_vector_type(2)));
typedef float v8f __attribute__((ext_vector_type(8)));

#define NEG_SLOPE 0.2f

// ---- monotonic float<->uint encoding so segment-max maps to GLOBAL_ATOMIC_MAX_U32 ----
__device__ __forceinline__ unsigned enc_f32(float f) {
  unsigned u = __float_as_uint(f);
  return (u & 0x80000000u) ? ~u : (u | 0x80000000u);
}
__device__ __forceinline__ float dec_f32(unsigned u) {
  return (u & 0x80000000u) ? __uint_as_float(u ^ 0x80000000u) : __uint_as_float(~u);
}

__device__ __forceinline__ float leaky(float x) { return x > 0.f ? x : NEG_SLOPE * x; }

__device__ __forceinline__ void edge_ends(const int* __restrict__ src,
                                          const int* __restrict__ dst,
                                          int e, int E, int& s, int& d) {
  if (e < E) { s = src[e]; d = dst[e]; } else { s = e - E; d = e - E; }  // self-loops
}

// =====================================================================
// H[n_rows x 128] = X[n_rows x 128] @ W[128 x 128]  (row-major, f32)
// One wave32 -> one 16x16 tile via V_WMMA_F32_16X16X4_F32, K stepped by 4.
// Block = 256 threads = 8 waves; wave w handles N-tile w (8 * 16 = 128 cols).
// =====================================================================
__global__ void gemm128_wmma_f32(const float* __restrict__ X,
                                 const float* __restrict__ W,
                                 float* __restrict__ H, int n_rows) {
  const int wave  = threadIdx.x >> 5;   // 0..7 -> N tile
  const int lane  = threadIdx.x & 31;
  const int lhalf = lane >> 4;          // 0: lanes 0-15, 1: lanes 16-31
  const int l15   = lane & 15;
  const int m0    = blockIdx.x * 16;
  const int n0    = wave * 16;

  int row  = m0 + l15;
  int arow = row < n_rows ? row : (n_rows - 1);   // clamp: EXEC must stay all-1s for WMMA
  const float* xrow = X + (size_t)arow * 128;

  v8f acc = {};
  #pragma unroll
  for (int k0 = 0; k0 < 128; k0 += 4) {
    const int ka = k0 + lhalf * 2;      // A: lanes0-15 hold K=k0,k0+1 ; lanes16-31 K=k0+2,k0+3
    v2f a, b;
    a.x = xrow[ka];
    a.y = xrow[ka + 1];
    b.x = W[(size_t)ka * 128 + n0 + l15];        // B: same half-split over K, lane&15 = col
    b.y = W[(size_t)(ka + 1) * 128 + n0 + l15];
    acc = __builtin_amdgcn_wmma_f32_16x16x4_f32(false, a, false, b, (short)0, acc,
                                                false, false);
  }
  // C/D layout: VGPR i -> M = i (lanes 0-15) / i+8 (lanes 16-31), N = lane&15
  #pragma unroll
  for (int i = 0; i < 8; ++i) {
    int r = m0 + lhalf * 8 + i;
    if (r < n_rows) H[(size_t)r * 128 + n0 + l15] = acc[i];
  }
}

// =====================================================================
// als[n,h] = <H[n, h*ch : (h+1)*ch], a_src[h]> ; ald likewise (one pass over H)
// =====================================================================
__global__ void attn_logits(const float* __restrict__ H,
                            const float* __restrict__ a_src,
                            const float* __restrict__ a_dst,
                            float* __restrict__ als, float* __restrict__ ald,
                            int n, int heads, int ch) {
  int i = blockIdx.x * blockDim.x + threadIdx.x;
  if (i >= n * heads) return;
  int node = i / heads, h = i % heads;
  const float* hp = H + (size_t)node * heads * ch + (size_t)h * ch;
  const float* as = a_src + h * ch;
  const float* ad = a_dst + h * ch;
  float s = 0.f, d = 0.f;
  for (int c = 0; c < ch; ++c) { float v = hp[c]; s = fmaf(v, as[c], s); d = fmaf(v, ad[c], d); }
  als[i] = s; ald[i] = d;
}

__global__ void init_softmax(unsigned* __restrict__ emax, float* __restrict__ denom, int n) {
  int i = blockIdx.x * blockDim.x + threadIdx.x;
  if (i < n) { emax[i] = enc_f32(-__builtin_inff()); denom[i] = 0.f; }
}

__global__ void init_out_bias(float* __restrict__ out, const float* __restrict__ bias,
                              int total, int cdim) {
  int i = blockIdx.x * blockDim.x + threadIdx.x;
  if (i < total) out[i] = bias[i % cdim];
}

// Pass A: segment max of leaky_relu(als[src]+ald[dst]) over dst (native u32 atomic max)
__global__ void edge_max(const int* __restrict__ src, const int* __restrict__ dst,
                         const float* __restrict__ als, const float* __restrict__ ald,
                         unsigned* __restrict__ emax, int E, int n, int heads) {
  int t = blockIdx.x * blockDim.x + threadIdx.x;
  int total = (E + n) * heads;
  if (t >= total) return;
  int e = t / heads, h = t % heads;
  int s, d; edge_ends(src, dst, e, E, s, d);
  float x = leaky(als[s * heads + h] + ald[d * heads + h]);
  atomicMax(&emax[d * heads + h], enc_f32(x));
}

// Pass B: alpha[e,h] = exp(e_logit - emax[dst]);  denom[dst] += alpha (one exp per edge-head)
__global__ void edge_exp(const int* __restrict__ src, const int* __restrict__ dst,
                         const float* __restrict__ als, const float* __restrict__ ald,
                         const unsigned* __restrict__ emax, float* __restrict__ denom,
                         float* __restrict__ alpha, int E, int n, int heads) {
  int t = blockIdx.x * blockDim.x + threadIdx.x;
  int total = (E + n) * heads;
  if (t >= total) return;
  int e = t / heads, h = t % heads;
  int s, d; edge_ends(src, dst, e, E, s, d);
  float x  = leaky(als[s * heads + h] + ald[d * heads + h]);
  float ex = expf(x - dec_f32(emax[d * heads + h]));
  alpha[t] = ex;
  atomicAdd(&denom[d * heads + h], ex);
}

// Pass B2: alpha[e,h] /= (denom[dst] + eps)
__global__ void edge_norm(const int* __restrict__ dst,
                          const float* __restrict__ denom,
                          float* __restrict__ alpha, int E, int n, int heads) {
  int t = blockIdx.x * blockDim.x + threadIdx.x;
  int total = (E + n) * heads;
  if (t >= total) return;
  int e = t / heads, h = t % heads;
  int d = (e < E) ? dst[e] : (e - E);
  alpha[t] = alpha[t] / (denom[d * heads + h] + 1e-16f);
}

// Pass C: out[dst, c] += H[src, c] * alpha[e, head(c)]
// One thread per (edge, 4-channel group); cdim = heads*ch = 128 -> 32 groups per edge.
// float4 gather of H (global_load_b128) + 4 native f32 atomic adds.
__global__ void edge_aggr_v4(const int* __restrict__ src, const int* __restrict__ dst,
                             const float* __restrict__ alpha,
                             const float* __restrict__ H, float* __restrict__ out,
                             int E, int n, int heads, int ch) {
  int t = blockIdx.x * blockDim.x + threadIdx.x;
  int total = (E + n) * 32;
  if (t >= total) return;
  int e = t >> 5;
  int c = (t & 31) << 2;           // channel base (0,4,...,124)
  int h = c / ch;
  int s, d; edge_ends(src, dst, e, E, s, d);
  float a = alpha[e * heads + h];
  const float4 hv = *(const float4*)(H + (size_t)s * 128 + c);
  float* op = out + (size_t)d * 128 + c;
  atomicAdd(op + 0, hv.x * a);
  atomicAdd(op + 1, hv.y * a);
  atomicAdd(op + 2, hv.z * a);
  atomicAdd(op + 3, hv.w * a);
}

__global__ void elu_inplace(float* __restrict__ x, int total) {
  int i = blockIdx.x * blockDim.x + threadIdx.x;
  if (i >= total) return;
  float v = x[i];
  x[i] = v > 0.f ? v : expf(v) - 1.f;
}

extern "C" void kernel_launch(void* const* d_in, const int* in_sizes, int n_in,
                              void* d_out, int out_size, void* d_ws, size_t ws_size,
                              hipStream_t stream) {
  const float* x      = (const float*)d_in[0];
  const int*   ei     = (const int*)d_in[1];
  const float* W1     = (const float*)d_in[2];
  const float* a_src1 = (const float*)d_in[3];
  const float* a_dst1 = (const float*)d_in[4];
  const float* b1     = (const float*)d_in[5];
  const float* W2     = (const float*)d_in[6];
  const float* a_src2 = (const float*)d_in[7];
  const float* a_dst2 = (const float*)d_in[8];
  const float* b2     = (const float*)d_in[9];
  float*       out    = (float*)d_out;

  const int N = in_sizes[0] / 128;
  const int E = in_sizes[1] / 2;
  const int* src = ei;
  const int* dst = ei + E;
  const int Etot = E + N;

  // workspace carve-up (~92 MB)
  float* p = (float*)d_ws;
  float* h1 = p;            p += (size_t)N * 128;
  float* x2 = p;            p += (size_t)N * 128;  // layer-1 out -> ELU -> layer-2 input
  float* h2 = p;            p += (size_t)N * 128;
  float* als1 = p;          p += (size_t)N * 4;
  float* ald1 = p;          p += (size_t)N * 4;
  unsigned* emax1 = (unsigned*)p; p += (size_t)N * 4;
  float* denom1 = p;        p += (size_t)N * 4;
  float* als2 = p;          p += N;
  float* ald2 = p;          p += N;
  unsigned* emax2 = (unsigned*)p; p += N;
  float* denom2 = p;        p += N;
  float* alpha = p;         p += (size_t)Etot * 4;  // reused by both layers

  const int B = 256;
  const int gN128 = (N * 128 + B - 1) / B;
  const int gE1   = (Etot * 4 + B - 1) / B;        // (edge, head) threads, layer 1
  const int gE2   = (Etot + B - 1) / B;            // layer 2 (heads=1)
  const int gEagg = (Etot * 32 + B - 1) / B;       // (edge, float4 group) threads

  // ---------------- Layer 1: GATConv(128 -> 4 x 32, concat) ----------------
  gemm128_wmma_f32<<<(N + 15) / 16, 256, 0, stream>>>(x, W1, h1, N);
  attn_logits<<<(N * 4 + B - 1) / B, B, 0, stream>>>(h1, a_src1, a_dst1, als1, ald1, N, 4, 32);
  init_softmax<<<(N * 4 + B - 1) / B, B, 0, stream>>>(emax1, denom1, N * 4);
  init_out_bias<<<gN128, B, 0, stream>>>(x2, b1, N * 128, 128);
  edge_max <<<gE1, B, 0, stream>>>(src, dst, als1, ald1, emax1, E, N, 4);
  edge_exp <<<gE1, B, 0, stream>>>(src, dst, als1, ald1, emax1, denom1, alpha, E, N, 4);
  edge_norm<<<gE1, B, 0, stream>>>(dst, denom1, alpha, E, N, 4);
  edge_aggr_v4<<<gEagg, B, 0, stream>>>(src, dst, alpha, h1, x2, E, N, 4, 32);
  elu_inplace<<<gN128, B, 0, stream>>>(x2, N * 128);

  // ---------------- Layer 2: GATConv(128 -> 128, heads=1) ----------------
  gemm128_wmma_f32<<<(N + 15) / 16, 256, 0, stream>>>(x2, W2, h2, N);
  attn_logits<<<(N + B - 1) / B, B, 0, stream>>>(h2, a_src2, a_dst2, als2, ald2, N, 1, 128);
  init_softmax<<<(N + B - 1) / B, B, 0, stream>>>(emax2, denom2, N);
  init_out_bias<<<gN128, B, 0, stream>>>(out, b2, N * 128, 128);
  edge_max <<<gE2, B, 0, stream>>>(src, dst, als2, ald2, emax2, E, N, 1);
  edge_exp <<<gE2, B, 0, stream>>>(src, dst, als2, ald2, emax2, denom2, alpha, E, N, 1);
  edge_norm<<<gE2, B, 0, stream>>>(dst, denom2, alpha, E, N, 1);
  edge_aggr_v4<<<gEagg, B, 0, stream>>>(src, dst, alpha, h2, out, E, N, 1, 128);
}